// Generator_73297911873809
// MI455X (gfx1250) — compile-verified
//
#include <hip/hip_runtime.h>
#include <hip/hip_bf16.h>

// MI455X (gfx1250) implementation.
// Pipeline:
//   K0: zero za scratch (64 x 32 fp32, nr padded 20->32)
//   K1: za = V * A^T via V_WMMA_F32_16X16X4_F32, split-K over 256 WGs,
//       bilinear upsample of x computed on the fly into LDS, A staged in LDS,
//       split-K reduced with global_atomic_add_f32.
//   K2: polynomial recurrence z (64 x 4), trivial.
//   K3: out[b,s] = dot(z[b,:], C[s,:]) + bias[s]  -- 67 MB coalesced NT stores.

typedef float v2f __attribute__((ext_vector_type(2)));
typedef float v4f __attribute__((ext_vector_type(4)));
typedef float v8f __attribute__((ext_vector_type(8)));

#define S_TOT   (512 * 512)     // 262144
#define BATCH   64
#define NR      20              // N_DEG * RANK
#define NR_PAD  32
#define KCHUNK  128
#define SLICE   1024            // K elements per workgroup
#define NBLK1   (S_TOT / SLICE) // 256 workgroups for phase 1

// ---------------------------------------------------------------- zero init
__global__ __launch_bounds__(256) void k0_zero(float* __restrict__ p, int n) {
    int i = blockIdx.x * 256 + threadIdx.x;
    if (i < n) p[i] = 0.0f;
}

// ---------------------------------------------------------------- phase 1: za GEMM (WMMA)
__global__ __launch_bounds__(256) void k1_za_wmma(const float* __restrict__ x,
                                                  const float* __restrict__ A,
                                                  float* __restrict__ za) {
    __shared__ float v_tile[BATCH][KCHUNK];   // 32 KB: upsampled values (A-operand rows)
    __shared__ float a_tile[NR_PAD][KCHUNK];  // 16 KB: A rows (B-operand cols), rows 20..31 = 0

    const int t    = threadIdx.x;
    const int lane = t & 31;
    const int w    = t >> 5;        // wave id 0..7 (wave32)
    const int mt   = w & 3;         // batch tile (16 rows each)
    const int nt   = w >> 2;        // nr tile (0..1)
    const int half = lane >> 4;     // 0 or 1 (K-pair selector)
    const int lmod = lane & 15;

    v8f acc = {};                   // fp32 16x16 accumulator (8 VGPRs)

    const int s_slice = blockIdx.x * SLICE;

    for (int chunk = 0; chunk < SLICE / KCHUNK; ++chunk) {
        const int s_base = s_slice + chunk * KCHUNK;
        // Chunks are 128-aligned and a dst row is 512 wide -> dst row constant per chunk.
        const int irow  = s_base >> 9;
        const int jbase = s_base & 511;

        // Vertical bilinear weights (half-pixel centers, scale 4): src = 0.25*i - 0.375
        const float fi  = 0.25f * (float)irow - 0.375f;
        const float fif = floorf(fi);
        const float wi  = fi - fif;
        int i0 = (int)fif;
        int i1 = i0 + 1;
        i0 = min(max(i0, 0), 127);
        i1 = min(max(i1, 0), 127);

        // Stage A slab: 32 rows x 128 (rows >= 20 zero-padded). Coalesced, streamed once.
        for (int e = t; e < NR_PAD * KCHUNK; e += 256) {
            const int row = e >> 7;
            const int sl  = e & (KCHUNK - 1);
            float val = 0.0f;
            if (row < NR)
                val = __builtin_nontemporal_load(&A[(size_t)row * S_TOT + (size_t)(s_base + sl)]);
            a_tile[row][sl] = val;
        }

        // Stage v slab: 64 x 128, bilinear upsample from x (x is hot in 192 MB L2).
        for (int e = t; e < BATCH * KCHUNK; e += 256) {
            const int b  = e >> 7;
            const int sl = e & (KCHUNK - 1);
            const int j  = jbase + sl;
            const float fj  = 0.25f * (float)j - 0.375f;
            const float fjf = floorf(fj);
            const float wj  = fj - fjf;
            int j0 = (int)fjf;
            int j1 = j0 + 1;
            j0 = min(max(j0, 0), 127);
            j1 = min(max(j1, 0), 127);
            const float* xb = x + (size_t)b * (128 * 128);
            const float x00 = xb[i0 * 128 + j0];
            const float x01 = xb[i0 * 128 + j1];
            const float x10 = xb[i1 * 128 + j0];
            const float x11 = xb[i1 * 128 + j1];
            const float top = x00 + wj * (x01 - x00);
            const float bot = x10 + wj * (x11 - x10);
            v_tile[b][sl] = top + wi * (bot - top);
        }
        __syncthreads();

        // WMMA inner loop: D(16x16) += A(16x4) x B(4x16), K-step 4.
        // A-operand lane layout: M = lane%16, K = {2*(lane/16), 2*(lane/16)+1}
        // B-operand lane layout: N = lane%16, same K pairing -> contiguous float2 LDS reads.
        #pragma unroll 8
        for (int k0 = 0; k0 < KCHUNK; k0 += 4) {
            const int kk = k0 + 2 * half;
            v2f a, bb;
            a.x  = v_tile[mt * 16 + lmod][kk];
            a.y  = v_tile[mt * 16 + lmod][kk + 1];
            bb.x = a_tile[nt * 16 + lmod][kk];
            bb.y = a_tile[nt * 16 + lmod][kk + 1];
            acc = __builtin_amdgcn_wmma_f32_16x16x4_f32(
                /*neg_a=*/false, a, /*neg_b=*/false, bb,
                /*c_mod=*/(short)0, acc, /*reuse_a=*/false, /*reuse_b=*/false);
        }
        __syncthreads();
    }

    // Split-K reduction: C/D layout VGPR i -> M = i + 8*(lane/16), N = lane%16.
    #pragma unroll
    for (int i = 0; i < 8; ++i) {
        const int bg = mt * 16 + i + 8 * half;
        const int nr = nt * 16 + lmod;
        if (nr < NR) atomicAdd(&za[bg * NR_PAD + nr], acc[i]);
    }
}

// ---------------------------------------------------------------- phase 2: polynomial recurrence
__global__ __launch_bounds__(256) void k2_poly(const float* __restrict__ za,
                                               float* __restrict__ z) {
    const int t = threadIdx.x;          // 256 threads == 64 b * 4 r
    const int b = t >> 2;
    const int r = t & 3;
    float zz = za[b * NR_PAD + r];      // n = 0
    #pragma unroll
    for (int n = 1; n < 5; ++n)
        zz = za[b * NR_PAD + n * 4 + r] * zz + zz;
    z[b * 4 + r] = zz;
}

// ---------------------------------------------------------------- phase 3: rank-4 expansion (67 MB out)
__global__ __launch_bounds__(256) void k3_out(const float* __restrict__ C,
                                              const float* __restrict__ bias,
                                              const float* __restrict__ z,
                                              float* __restrict__ out) {
    __shared__ float zs[BATCH * 4];
    const int t = threadIdx.x;
    zs[t] = z[t];                       // 256 floats exactly
    __syncthreads();

    const int s = blockIdx.x * 256 + t;
    const v4f c4 = __builtin_nontemporal_load(&((const v4f*)C)[s]); // coalesced b128
    const float bv = bias[s];

    #pragma unroll 4
    for (int b = 0; b < BATCH; ++b) {
        const float* zb = &zs[b * 4];
        float acc = fmaf(zb[0], c4.x, bv);
        acc = fmaf(zb[1], c4.y, acc);
        acc = fmaf(zb[2], c4.z, acc);
        acc = fmaf(zb[3], c4.w, acc);
        __builtin_nontemporal_store(acc, &out[(size_t)b * S_TOT + s]);
    }
}

// ---------------------------------------------------------------- launcher
extern "C" void kernel_launch(void* const* d_in, const int* in_sizes, int n_in,
                              void* d_out, int out_size, void* d_ws, size_t ws_size,
                              hipStream_t stream) {
    const float* x  = (const float*)d_in[0];   // (64,1,128,128)
    const float* A  = (const float*)d_in[1];   // (5,4,S)
    const float* C  = (const float*)d_in[2];   // (S,4)
    const float* bb = (const float*)d_in[3];   // (S,)
    float* out = (float*)d_out;                // (64,1,512,512)

    float* za = (float*)d_ws;                  // 64*32 fp32
    float* z  = za + BATCH * NR_PAD;           // 64*4 fp32

    k0_zero<<<(BATCH * NR_PAD + 255) / 256, 256, 0, stream>>>(za, BATCH * NR_PAD);
    k1_za_wmma<<<NBLK1, 256, 0, stream>>>(x, A, za);
    k2_poly<<<1, 256, 0, stream>>>(za, z);
    k3_out<<<S_TOT / 256, 256, 0, stream>>>(C, bb, z, out);
}